// DotProcessorBlock_88476326298187
// MI455X (gfx1250) — compile-verified
//
#include <hip/hip_runtime.h>

typedef __attribute__((ext_vector_type(2))) float v2f;
typedef __attribute__((ext_vector_type(8))) float v8f;

#define NFEAT 256
#define NTRI  32896   // 256*257/2 = 128*256 + 128

// One block per batch row. 256 threads = 8 waves (wave32).
// Wave w computes row-tile i0 = 16*w of the 256x256 outer product via
// V_WMMA_F32_16X16X4_F32 rank-1 updates, sweeping 16 column tiles.
// Rows 0..127 fully covered by the 8x16 tile grid; row 128 (first 128 cols)
// handled by a scalar epilogue.
__global__ __launch_bounds__(256) void outer_tri_wmma_kernel(
    const float* __restrict__ x,
    const float* __restrict__ w,
    const float* __restrict__ bias,
    float* __restrict__ out)
{
    __shared__ float feat[NFEAT];

    const int tid = threadIdx.x;
    const int b   = blockIdx.x;

    // feat = x * w + b for this batch row (exact f32, matches reference)
    feat[tid] = fmaf(x[(size_t)b * NFEAT + tid], w[tid], bias[tid]);
    __syncthreads();

    const int wave = tid >> 5;      // 0..7 -> row-tile
    const int lane = tid & 31;
    const int half = lane >> 4;     // 0: lanes 0-15, 1: lanes 16-31
    const int n    = lane & 15;

    const int i0 = wave * 16;

    // A fragment (16x4 f32, 2 VGPRs/lane):
    //   lanes 0-15 : VGPR0 = K0 (row m = lane), VGPR1 = K1
    //   lanes 16-31: VGPR0 = K2,                VGPR1 = K3
    // We want A[m][k] = feat[i0+m] for k==0, else 0  ->  rank-1 product.
    v2f a;
    a.x = (half == 0) ? feat[i0 + n] : 0.0f;
    a.y = 0.0f;

    float* outb = out + (size_t)b * NTRI;

    for (int tj = 0; tj < 16; ++tj) {
        const int j0 = tj * 16;
        // B fragment (4x16 f32): replicate feat[j0+n] across all K rows so the
        // result does not depend on B's exact K<->lane-half mapping (only A's
        // K=0 column is nonzero).
        const float bv = feat[j0 + n];
        v2f bb; bb.x = bv; bb.y = bv;

        v8f c = {};
        // D = A x B + 0 : D[m][n] = feat[i0+m] * feat[j0+n] (exact f32)
        c = __builtin_amdgcn_wmma_f32_16x16x4_f32(
                /*neg_a=*/false, a, /*neg_b=*/false, bb,
                /*c_mod=*/(short)0, c, /*reuse_a=*/false, /*reuse_b=*/false);

        // C/D layout: VGPR r -> M = i0 + r + 8*half, N = j0 + n.
        // Output is the first 32896 entries of row-major 256x256:
        // element (i,j) lives at i*256 + j.
        #pragma unroll
        for (int r = 0; r < 8; ++r) {
            const int row = i0 + r + half * 8;
            __builtin_nontemporal_store(c[r], &outb[row * NFEAT + j0 + n]);
        }
    }

    // Tail: row i = 128, columns 0..127 (indices 32768..32895).
    if (tid < 128) {
        __builtin_nontemporal_store(feat[128] * feat[tid],
                                    &outb[128 * NFEAT + tid]);
    }
}

extern "C" void kernel_launch(void* const* d_in, const int* in_sizes, int n_in,
                              void* d_out, int out_size, void* d_ws, size_t ws_size,
                              hipStream_t stream) {
    const float* x    = (const float*)d_in[0];   // (B, 256)
    const float* ww   = (const float*)d_in[1];   // (256,)
    const float* bias = (const float*)d_in[2];   // (256,)
    float* out        = (float*)d_out;           // (B, 32896)

    const int B = in_sizes[0] / NFEAT;           // 4096
    outer_tri_wmma_kernel<<<B, 256, 0, stream>>>(x, ww, bias, out);
}